// fdlp_spectrogram_18588618457710
// MI455X (gfx1250) — compile-verified
//
#include <hip/hip_runtime.h>

typedef float v2f __attribute__((ext_vector_type(2)));
typedef float v4f __attribute__((ext_vector_type(4)));
typedef float v8f __attribute__((ext_vector_type(8)));

#define FLEN   24000
#define NFRM   12
#define NBAT   8
#define NBF    96      // NBAT*NFRM = GEMM N dimension
#define NFILT  20
#define ORDER  80
#define NCOEF  80
#define CUT    188
#define CUTOV  94
#define TOUT   1222
#define HOP    12000
#define TLEN   160000
#define PHMOD  96000   // 4*FLEN : phase modulus in units of pi/(2*FLEN) (revolutions = m/96000)
#define CHUNK  128
#define NTASK  (NBF * NFILT)   // 1920

__device__ __forceinline__ float wred32(float v) {
  v += __shfl_xor(v, 16, 32);
  v += __shfl_xor(v, 8, 32);
  v += __shfl_xor(v, 4, 32);
  v += __shfl_xor(v, 2, 32);
  v += __shfl_xor(v, 1, 32);
  return v;
}

// ---------------------------------------------------------------------------
// Stage 1: DCT-II as GEMM  mod[bf][k] = 2 * sum_n cos(pi*k*(2n+1)/(2*FLEN)) * x[n,bf]
// A (16x4 f32) generated on the fly via v_cos (revolutions), B staged in LDS.
// One wave -> 16 k-rows x 96 frame-columns (6 WMMA tiles). 8 waves/WG -> 128 k rows.
// ---------------------------------------------------------------------------
__global__ __launch_bounds__(256) void fdlp_dct_gemm(const float* __restrict__ speech,
                                                     float* __restrict__ mod) {
  __shared__ float xs[CHUNK][NBF];
  const int tid  = threadIdx.x;
  const int lane = tid & 31;
  const int wv   = tid >> 5;
  const int k0w  = blockIdx.x * 128 + wv * 16;
  const int krow = k0w + (lane & 15);
  const int kbase = (lane >> 4) * 2;         // A layout: V0 holds K=0 (lanes0-15) / K=2 (lanes16-31)

  // incremental integer phase: m = krow*(2*(n+kbase)+1) mod 96000
  unsigned m     = (unsigned)(((long long)krow * (2 * kbase + 1)) % PHMOD);
  const unsigned step4 = (unsigned)(((long long)krow * 8) % PHMOD);   // n += 4
  const unsigned step1 = (unsigned)(((long long)krow * 2) % PHMOD);   // n += 1 (for a.y)

  v8f acc[6] = {};                           // 6 tiles of 16x16 f32 accumulators
  const int bN = lane & 15;                  // B/C column within tile
  const int kb = lane >> 4;                  // B layout: V0 -> K = 0/1, V1 -> K = 2/3

  for (int n0 = 0; n0 < FLEN; n0 += CHUNK) {
    const int cn = min(CHUNK, FLEN - n0);    // 128 or 64, always multiple of 4
    __syncthreads();
    for (int e = tid; e < cn * NBF; e += 256) {
      const int nl = e / NBF;
      const int bf = e - nl * NBF;
      const int b  = bf / NFRM;
      const int fr = bf - b * NFRM;
      xs[nl][bf] = speech[b * TLEN + fr * HOP + n0 + nl];
    }
    __syncthreads();
    for (int nn = 0; nn < cn; nn += 4) {
      // A fragment: cos in revolutions (v_cos_f32 input = radians/2pi), exact int phase
      unsigned m2 = m + step1; if (m2 >= PHMOD) m2 -= PHMOD;
      v2f a;
      a.x = __builtin_amdgcn_cosf((float)m  * (1.0f / PHMOD));
      a.y = __builtin_amdgcn_cosf((float)m2 * (1.0f / PHMOD));
      m += step4; if (m >= PHMOD) m -= PHMOD;
#pragma unroll
      for (int j = 0; j < 6; ++j) {
        v2f bfrag;
        bfrag.x = xs[nn + kb][j * 16 + bN];
        bfrag.y = xs[nn + kb + 2][j * 16 + bN];
        acc[j] = __builtin_amdgcn_wmma_f32_16x16x4_f32(
            false, a, false, bfrag, (short)0, acc[j], false, false);
      }
    }
  }

  // C/D layout: lane l -> column N=l%16 ; VGPR v -> row M = v + 8*(l/16)
  if (k0w + 16 <= FLEN) {
    const int kk = k0w + 8 * (lane >> 4);
#pragma unroll
    for (int j = 0; j < 6; ++j) {
      const int bf = j * 16 + bN;
      float* dst = mod + (size_t)bf * FLEN + kk;
      v4f lo, hi;
#pragma unroll
      for (int v = 0; v < 4; ++v) { lo[v] = 2.0f * acc[j][v]; hi[v] = 2.0f * acc[j][v + 4]; }
      *(v4f*)(dst)     = lo;
      *(v4f*)(dst + 4) = hi;
    }
  }
}

// ---------------------------------------------------------------------------
// Stage 2: per (bf, filt): y = mod*fbank, circular autocorr lags 0..80.
// Full 24000-float y lives in LDS (96 KB of the 320 KB WGP LDS).
// ---------------------------------------------------------------------------
__global__ __launch_bounds__(256) void fdlp_autocorr(const float* __restrict__ mod,
                                                     const float* __restrict__ fbank,
                                                     float* __restrict__ R) {
  extern __shared__ float ybuf[];
  const int blk  = blockIdx.x;           // bf*NFILT + filt
  const int bf   = blk / NFILT;
  const int filt = blk - bf * NFILT;
  const float* mr = mod + (size_t)bf * FLEN;
  const float* fb = fbank + (size_t)filt * FLEN;
  const int tid = threadIdx.x;
  for (int i = tid; i < FLEN; i += 256) {
    __builtin_prefetch(mr + i + 4096, 0, 0);
    ybuf[i] = mr[i] * fb[i];
  }
  __syncthreads();
  const int lane = tid & 31, wv = tid >> 5;
  for (int lag = wv; lag <= ORDER; lag += 8) {
    float acc = 0.f;
    for (int n = lane; n < FLEN; n += 32) {
      int n2 = n + lag; n2 = (n2 >= FLEN) ? n2 - FLEN : n2;
      acc = fmaf(ybuf[n], ybuf[n2], acc);
    }
    acc = wred32(acc);
    if (lane == 0) R[(size_t)blk * (ORDER + 1) + lag] = acc;
  }
}

// ---------------------------------------------------------------------------
// Stage 3: Levinson-Durbin(80) -> LPC -> cepstrum(80) -> 188-pt envelope.
// One wave32 per task; per-task arrays in LDS; cross-lane reduce via shuffles.
// ---------------------------------------------------------------------------
__global__ __launch_bounds__(256) void fdlp_lev_cep_env(const float* __restrict__ R,
                                                        const float* __restrict__ lifter,
                                                        const float* __restrict__ mask,
                                                        float* __restrict__ env) {
  __shared__ float sl[8][256];   // per wave: R[0..80] @0, a[0..79] @88, c[0..79] @176
  const int tid = threadIdx.x, lane = tid & 31, wv = tid >> 5;
  const int task = blockIdx.x * 8 + wv;          // < 1920 exactly
  float* Rl = &sl[wv][0];
  float* al = &sl[wv][88];
  float* cl = &sl[wv][176];
  const float* Rg = R + (size_t)task * (ORDER + 1);
  for (int i = lane; i <= ORDER; i += 32) Rl[i] = Rg[i];
  for (int j = lane; j < ORDER; j += 32) al[j] = 0.f;
  asm volatile("s_wait_dscnt 0x0" ::: "memory");

  // ---- Levinson-Durbin ----
  float err = Rl[0];
  for (int i = 1; i <= ORDER; ++i) {
    float part = 0.f;
    for (int j = lane; j <= i - 2; j += 32) part = fmaf(al[j], Rl[i - 1 - j], part);
    const float num = Rl[i] - wred32(part);
    const float k   = num / err;
    float ov[3], orv[3];
    int c = 0;
    for (int j = lane; j <= i - 2; j += 32) { ov[c] = al[j]; orv[c] = al[i - 2 - j]; ++c; }
    c = 0;
    for (int j = lane; j <= i - 2; j += 32) { al[j] = fmaf(-k, orv[c], ov[c]); ++c; }
    if (lane == 0) al[i - 1] = k;
    asm volatile("s_wait_dscnt 0x0" ::: "memory");
    err = (1.0f - k * k) * err;
  }
  const float gain = err;

  // a = lpc[1:] = -alpha
  for (int j = lane; j < ORDER; j += 32) al[j] = -al[j];
  if (lane == 0) cl[0] = 0.5f * __builtin_amdgcn_logf(gain + 1e-12f) * 0.6931471805599453f;
  asm volatile("s_wait_dscnt 0x0" ::: "memory");

  // ---- cepstrum recursion ----
  for (int n = 1; n < NCOEF; ++n) {
    float part = 0.f;
    const float invn = 1.0f / (float)n;
    for (int k = lane; k <= n - 1; k += 32)
      if (k >= 1) part = fmaf((float)k * invn * cl[k], al[n - k - 1], part);
    const float accv = wred32(part);
    if (lane == 0) cl[n] = -(al[n - 1] + accv);
    asm volatile("s_wait_dscnt 0x0" ::: "memory");
  }

  // liftering (mask zeroes c[0])
  for (int mm = lane; mm < NCOEF; mm += 32) cl[mm] *= lifter[mm] * mask[mm];
  asm volatile("s_wait_dscnt 0x0" ::: "memory");

  // ---- envelope: env[t] = exp(2 * sum_m c[m] * cos(2*pi*t*m/376)), t < 188 ----
  float* eo = env + (size_t)task * CUT;
  for (int t = lane; t < CUT; t += 32) {
    float s = 0.f;
    for (int mm = 0; mm < NCOEF; ++mm) {
      const int ph = (t * mm) % (2 * CUT);           // revolutions numerator
      s = fmaf(cl[mm], __builtin_amdgcn_cosf((float)ph * (1.0f / (2 * CUT))), s);
    }
    eo[t] = __builtin_amdgcn_exp2f(s * (2.0f * 1.4426950408889634f));
  }
}

// ---------------------------------------------------------------------------
// Stage 4: overlap-add (gather form, 1-2 frames per t), counts, log.
// ---------------------------------------------------------------------------
__global__ __launch_bounds__(256) void fdlp_finalize(const float* __restrict__ env,
                                                     float* __restrict__ out) {
  const int idx = blockIdx.x * 256 + threadIdx.x;
  const int total = NBAT * TOUT * NFILT;
  if (idx >= total) return;
  const int filt = idx % NFILT;
  const int rem  = idx / NFILT;
  const int t    = rem % TOUT;
  const int b    = rem / TOUT;
  int frHi = t / CUTOV; if (frHi > NFRM - 1) frHi = NFRM - 1;
  const int frLo = (t >= CUT) ? ((t - CUT) / CUTOV + 1) : 0;
  float s = 0.f; int cnt = 0;
  for (int fr = frLo; fr <= frHi; ++fr) {
    const int tt = t - fr * CUTOV;     // guaranteed in [0, CUT)
    s += env[((size_t)(b * NFRM + fr) * NFILT + filt) * CUT + tt];
    ++cnt;
  }
  out[idx] = __builtin_amdgcn_logf(s / (float)cnt + 1e-8f) * 0.6931471805599453f;
}

// ---------------------------------------------------------------------------
extern "C" void kernel_launch(void* const* d_in, const int* in_sizes, int n_in,
                              void* d_out, int out_size, void* d_ws, size_t ws_size,
                              hipStream_t stream) {
  const float* speech = (const float*)d_in[0];   // (8, 160000)
  const float* fbank  = (const float*)d_in[1];   // (20, 24000)
  const float* lifter = (const float*)d_in[2];   // (80,)
  const float* maskv  = (const float*)d_in[3];   // (80,)
  float* out = (float*)d_out;                    // (8, 1222, 20)

  float* ws   = (float*)d_ws;
  float* mod  = ws;                                        // 96 * 24000      = 2,304,000 f
  float* Rbuf = mod + (size_t)NBF * FLEN;                  // 1920 * 81       =   155,520 f
  float* envb = Rbuf + (size_t)NTASK * (ORDER + 1);        // 1920 * 188      =   360,960 f

  fdlp_dct_gemm<<<188, 256, 0, stream>>>(speech, mod);
  fdlp_autocorr<<<NTASK, 256, FLEN * sizeof(float), stream>>>(mod, fbank, Rbuf);
  fdlp_lev_cep_env<<<NTASK / 8, 256, 0, stream>>>(Rbuf, lifter, maskv, envb);
  fdlp_finalize<<<(NBAT * TOUT * NFILT + 255) / 256, 256, 0, stream>>>(envb, out);
}